// Localizer_47339129536976
// MI455X (gfx1250) — compile-verified
//
#include <hip/hip_runtime.h>
#include <math.h>

#define EPSF 1e-7f
#define INV_PI 0.31830988618379067154f

__device__ __forceinline__ float clamp1(float v) {
    return fminf(1.0f, fmaxf(-1.0f, v));
}

// Async DMA store: 8 bytes LDS -> global, tracked by ASYNCcnt (gfx1250).
// Operand order mirrors global_store_b64: vaddr(64b global), vdata(=LDS byte addr), saddr.
__device__ __forceinline__ void async_store_lds_b64(const float* gptr, const float* lptr) {
    unsigned long long g = (unsigned long long)(size_t)gptr;
    unsigned l = (unsigned)(size_t)lptr;   // low 32 bits of generic shared ptr == LDS offset
    asm volatile("global_store_async_from_lds_b64 %0, %1, off"
                 :: "v"(g), "v"(l) : "memory");
}

// Build node rotation frame R (row-major 3x3) and canon_vel = R^T v
// without transcendentals:
//   theta = atan2(vy,vx): ct = vx/rxy, st = vy/rxy  (rxy==0 -> ct=1,st=0)
//   phi   = acos(clip(vz/(rho+eps))): cp = clipped arg, sp = sqrt(1-cp^2) >= 0
__device__ __forceinline__ void node_frame(float vx, float vy, float vz,
                                           float R[9], float c[3]) {
    float rho = sqrtf(vx * vx + vy * vy + vz * vz);
    float cp  = clamp1(vz / (rho + EPSF));
    float sp  = sqrtf(fmaxf(0.0f, 1.0f - cp * cp));
    float rxy = sqrtf(vx * vx + vy * vy);
    float ct, st;
    if (rxy > 0.0f) {
        float inv = 1.0f / rxy;
        ct = vx * inv;
        st = vy * inv;
    } else {            // atan2(0,0) == 0
        ct = 1.0f;
        st = 0.0f;
    }
    R[0] = cp * ct; R[1] = -st;  R[2] = sp * ct;
    R[3] = cp * st; R[4] =  ct;  R[5] = sp * st;
    R[6] = -sp;     R[7] = 0.f;  R[8] = cp;
    // canon = R^T v
    c[0] = R[0] * vx + R[3] * vy + R[6] * vz;
    c[1] = R[1] * vx + R[4] * vy + R[7] * vz;
    c[2] = R[2] * vx + R[5] * vy + R[8] * vz;
}

// Phase 1: one thread per node. Writes rel_feat[N,6], Rn[N,3,3] and (optionally)
// a 48B-aligned packed table {R[9], canon[3]} for fast b128 gathers in phase 2.
__global__ __launch_bounds__(256) void node_kernel(
    const float* __restrict__ x,
    float* __restrict__ rel_feat,
    float* __restrict__ Rn,
    float* __restrict__ packed,     // may be null
    int N) {
    int i = blockIdx.x * blockDim.x + threadIdx.x;
    if (i >= N) return;
    const float* row = x + (size_t)i * 6;
    float vx = row[3], vy = row[4], vz = row[5];
    float R[9], c[3];
    node_frame(vx, vy, vz, R, c);

    float* rf = rel_feat + (size_t)i * 6;
    rf[0] = 0.0f; rf[1] = 0.0f; rf[2] = 0.0f;
    rf[3] = c[0]; rf[4] = c[1]; rf[5] = c[2];

    float* rr = Rn + (size_t)i * 9;
#pragma unroll
    for (int k = 0; k < 9; ++k) rr[k] = R[k];

    if (packed) {
        float4* pk = (float4*)(packed + (size_t)i * 12);   // 48B aligned
        pk[0] = make_float4(R[0], R[1], R[2], R[3]);
        pk[1] = make_float4(R[4], R[5], R[6], R[7]);
        pk[2] = make_float4(R[8], c[0], c[1], c[2]);
    }
}

// Phase 2: one thread per edge; 18-float rows staged through LDS, then drained
// to global with async LDS->memory DMA stores (ASYNCcnt path).
__global__ __launch_bounds__(256) void edge_kernel(
    const float* __restrict__ x,
    const int*   __restrict__ send,
    const int*   __restrict__ recv,
    const float* __restrict__ rel_feat,
    const float* __restrict__ Rn,
    const float* __restrict__ packed,   // may be null
    float* __restrict__ edge_attr,
    int E) {
    __shared__ float lds[256 * 18];

    const int tid = threadIdx.x;
    const long long blockStart = (long long)blockIdx.x * 256;
    const long long e = blockStart + tid;
    const int count = (int)((blockStart + 256 <= (long long)E)
                                ? 256
                                : ((long long)E - blockStart));

    if (e < (long long)E) {
        // gfx1250 global_prefetch_b8 on the streaming index arrays
        __builtin_prefetch(send + e + 4096, 0, 1);
        __builtin_prefetch(recv + e + 4096, 0, 1);

        const int j = send[e];
        const int i = recv[e];

        const float* xj = x + (size_t)j * 6;
        const float* xi = x + (size_t)i * 6;
        float pjx = xj[0], pjy = xj[1], pjz = xj[2];
        float vjx = xj[3], vjy = xj[4], vjz = xj[5];
        float pix = xi[0], piy = xi[1], piz = xi[2];

        float Ri[9], Rj[9], ci0, ci1, ci2;
        if (packed) {   // uniform branch; 3x b128 gathers per endpoint
            const float4* pi4 = (const float4*)(packed + (size_t)i * 12);
            float4 A = pi4[0], B = pi4[1], C = pi4[2];
            Ri[0] = A.x; Ri[1] = A.y; Ri[2] = A.z; Ri[3] = A.w;
            Ri[4] = B.x; Ri[5] = B.y; Ri[6] = B.z; Ri[7] = B.w;
            Ri[8] = C.x; ci0 = C.y; ci1 = C.z; ci2 = C.w;
            const float4* pj4 = (const float4*)(packed + (size_t)j * 12);
            float4 D = pj4[0], F = pj4[1], G = pj4[2];
            Rj[0] = D.x; Rj[1] = D.y; Rj[2] = D.z; Rj[3] = D.w;
            Rj[4] = F.x; Rj[5] = F.y; Rj[6] = F.z; Rj[7] = F.w;
            Rj[8] = G.x;
        } else {
            const float* pri = Rn + (size_t)i * 9;
            const float* prj = Rn + (size_t)j * 9;
#pragma unroll
            for (int k = 0; k < 9; ++k) { Ri[k] = pri[k]; Rj[k] = prj[k]; }
            const float* ci = rel_feat + (size_t)i * 6 + 3;
            ci0 = ci[0]; ci1 = ci[1]; ci2 = ci[2];
        }

        // rel_pos and rot_rel_pos = Ri^T rel_pos
        float rx = pjx - pix, ry = pjy - piy, rz = pjz - piz;
        float rr0 = Ri[0] * rx + Ri[3] * ry + Ri[6] * rz;
        float rr1 = Ri[1] * rx + Ri[4] * ry + Ri[7] * rz;
        float rr2 = Ri[2] * rx + Ri[5] * ry + Ri[8] * rz;

        // Needed entries of O = Ri^T Rj
        float O00 = Ri[0] * Rj[0] + Ri[3] * Rj[3] + Ri[6] * Rj[6];
        float O10 = Ri[1] * Rj[0] + Ri[4] * Rj[3] + Ri[7] * Rj[6];
        float O20 = Ri[2] * Rj[0] + Ri[5] * Rj[3] + Ri[8] * Rj[6];
        float O21 = Ri[2] * Rj[1] + Ri[5] * Rj[4] + Ri[8] * Rj[7];
        float O22 = Ri[2] * Rj[2] + Ri[5] * Rj[5] + Ri[8] * Rj[8];

        float eu0 = atan2f(O10, O00) * INV_PI;
        float eu1 = asinf(clamp1(-O20)) * INV_PI;
        float eu2 = atan2f(O21, O22) * INV_PI;

        float dist = sqrtf(rx * rx + ry * ry + rz * rz);

        float rho2 = sqrtf(rr0 * rr0 + rr1 * rr1 + rr2 * rr2);
        float th = atan2f(rr1, rr0);                        // symmetric theta
        float ph = acosf(clamp1(rr2 / (rho2 + EPSF)));

        // rot_vel = Ri^T vel_j
        float rv0 = Ri[0] * vjx + Ri[3] * vjy + Ri[6] * vjz;
        float rv1 = Ri[1] * vjx + Ri[4] * vjy + Ri[7] * vjz;
        float rv2 = Ri[2] * vjx + Ri[5] * vjy + Ri[8] * vjz;

        float* o = lds + tid * 18;
        o[0]  = rr0; o[1]  = rr1; o[2]  = rr2;
        o[3]  = eu0; o[4]  = eu1; o[5]  = eu2;
        o[6]  = dist;
        o[7]  = th;  o[8]  = ph;
        o[9]  = rv0; o[10] = rv1; o[11] = rv2;
        o[12] = 0.0f; o[13] = 0.0f; o[14] = 0.0f;
        o[15] = ci0; o[16] = ci1; o[17] = ci2;
    }
    __syncthreads();   // drains DScnt; all lanes' LDS rows visible

    // DMA-drain the block's contiguous span: count*72 bytes = count*9 b64 chunks.
    const int chunks = count * 9;
    float* outBase = edge_attr + blockStart * 18;
#pragma unroll
    for (int k = 0; k < 9; ++k) {
        int c = tid + k * 256;
        if (c < chunks) {
            async_store_lds_b64(outBase + (size_t)c * 2, lds + (size_t)c * 2);
        }
    }
    // Drain ASYNCcnt before wave retires (s_endpgm also implies wait-idle).
    asm volatile("s_wait_asynccnt 0" ::: "memory");
}

extern "C" void kernel_launch(void* const* d_in, const int* in_sizes, int n_in,
                              void* d_out, int out_size, void* d_ws, size_t ws_size,
                              hipStream_t stream) {
    (void)n_in; (void)out_size;
    const float* x    = (const float*)d_in[0];
    const int*   send = (const int*)d_in[1];
    const int*   recv = (const int*)d_in[2];

    const int N = in_sizes[0] / 6;
    const int E = in_sizes[1];

    float* out       = (float*)d_out;
    float* rel_feat  = out;                                   // [N,6]
    float* Rn        = out + (size_t)N * 6;                   // [N,3,3]
    float* edge_attr = out + (size_t)N * 6 + (size_t)N * 9;   // [E,18]

    // Packed {R[9],canon[3]} table in workspace if it fits (48B per node).
    float* packed = (ws_size >= (size_t)N * 48) ? (float*)d_ws : nullptr;

    node_kernel<<<(N + 255) / 256, 256, 0, stream>>>(x, rel_feat, Rn, packed, N);
    edge_kernel<<<(E + 255) / 256, 256, 0, stream>>>(x, send, recv,
                                                     rel_feat, Rn, packed,
                                                     edge_attr, E);
}